// GCNBlock_3143916060941
// MI455X (gfx1250) — compile-verified
//
#include <hip/hip_runtime.h>

typedef float v2f __attribute__((ext_vector_type(2)));
typedef float v8f __attribute__((ext_vector_type(8)));

#define FEAT 128

// ---------------------------------------------------------------------------
// Dual GEMM: H1 = X @ W1, H2 = X @ W2   (X:[M,128], W:[128,128] row-major)
// One wave per 16x16 output tile; shared A tile feeds both accumulators.
// Uses V_WMMA_F32_16X16X4_F32 (exact fp32, matches reference numerics).
// ---------------------------------------------------------------------------
__global__ __launch_bounds__(256)
void gcn_gemm_dual_wmma(const float* __restrict__ X,
                        const float* __restrict__ W1,
                        const float* __restrict__ W2,
                        float* __restrict__ H1,
                        float* __restrict__ H2,
                        int M) {
  const int lane = threadIdx.x & 31;
  const int wave = threadIdx.x >> 5;      // 8 waves -> 8 N-tiles (FEAT=128)
  const int m0 = blockIdx.x * 16;
  const int n0 = wave * 16;
  const int l15 = lane & 15;
  const int hi  = lane >> 4;              // 0: K pair {k,k+1}; 1: {k+2,k+3}

  int mload = m0 + l15;
  if (mload > M - 1) mload = M - 1;       // clamp A loads (tail tiles only)
  const float* __restrict__ xrow = X + (size_t)mload * FEAT;

  v8f c1 = {0.f, 0.f, 0.f, 0.f, 0.f, 0.f, 0.f, 0.f};
  v8f c2 = c1;

#pragma unroll 4
  for (int k = 0; k < FEAT; k += 4) {
    const int ka = k + 2 * hi;
    v2f a;  a.x  = xrow[ka];                      a.y  = xrow[ka + 1];
    v2f b1; b1.x = W1[ka * FEAT + n0 + l15];      b1.y = W1[(ka + 1) * FEAT + n0 + l15];
    v2f b2; b2.x = W2[ka * FEAT + n0 + l15];      b2.y = W2[(ka + 1) * FEAT + n0 + l15];
    c1 = __builtin_amdgcn_wmma_f32_16x16x4_f32(false, a, false, b1, (short)0, c1, false, false);
    c2 = __builtin_amdgcn_wmma_f32_16x16x4_f32(false, a, false, b2, (short)0, c2, false, false);
  }

  // C/D layout: VGPR r -> M = m0 + r + 8*hi, N = n0 + l15.
  const size_t base = (size_t)(m0 + 8 * hi) * FEAT + (n0 + l15);
  float* __restrict__ o1 = H1 + base;
  float* __restrict__ o2 = H2 + base;
  if (m0 + 16 <= M) {
    // Full tile (always true when M % 16 == 0): straight-line stores,
    // row stride 512 B folds into the instruction immediate offset.
#pragma unroll
    for (int r = 0; r < 8; ++r) {
      o1[(size_t)r * FEAT] = c1[r];
      o2[(size_t)r * FEAT] = c2[r];
    }
  } else {
#pragma unroll
    for (int r = 0; r < 8; ++r) {
      if (m0 + r + 8 * hi < M) {
        o1[(size_t)r * FEAT] = c1[r];
        o2[(size_t)r * FEAT] = c2[r];
      }
    }
  }
}

// ---------------------------------------------------------------------------
// Degree / normalization helpers.  Self-loop => every node starts at deg=1.
// ---------------------------------------------------------------------------
__global__ void deg_init(float* __restrict__ d1, float* __restrict__ d2, int n) {
  int i = blockIdx.x * blockDim.x + threadIdx.x;
  if (i < n) { d1[i] = 1.0f; d2[i] = 1.0f; }
}

__global__ void deg_count(const int* __restrict__ dst, float* __restrict__ deg, int e) {
  int i = blockIdx.x * blockDim.x + threadIdx.x;
  if (i < e) atomicAdd(&deg[dst[i]], 1.0f);
}

__global__ void deg_rsqrt(float* __restrict__ d1, float* __restrict__ d2, int n) {
  int i = blockIdx.x * blockDim.x + threadIdx.x;
  if (i < n) { d1[i] = rsqrtf(d1[i]); d2[i] = rsqrtf(d2[i]); }
}

__global__ void zero_f4(float4* __restrict__ p, size_t n4) {
  size_t i = (size_t)blockIdx.x * blockDim.x + threadIdx.x;
  if (i < n4) p[i] = make_float4(0.f, 0.f, 0.f, 0.f);
}

// ---------------------------------------------------------------------------
// Edge scatter: one wave per edge, 4 feats/lane (float4 gather + f32 atomics).
// norm computed on the fly: dinv[src]*dinv[dst].
// ---------------------------------------------------------------------------
__global__ __launch_bounds__(256)
void edge_scatter(const float* __restrict__ H,
                  const int* __restrict__ src,
                  const int* __restrict__ dst,
                  const float* __restrict__ dinv,
                  float* __restrict__ AGG,
                  int E) {
  const int e = blockIdx.x * 8 + (threadIdx.x >> 5);
  if (e >= E) return;
  const int lane = threadIdx.x & 31;
  const int s = src[e];
  const int d = dst[e];
  const float nrm = dinv[s] * dinv[d];
  const float4 v = *((const float4*)(H + (size_t)s * FEAT) + lane);
  float* ap = AGG + (size_t)d * FEAT + lane * 4;
  atomicAdd(ap + 0, v.x * nrm);
  atomicAdd(ap + 1, v.y * nrm);
  atomicAdd(ap + 2, v.z * nrm);
  atomicAdd(ap + 3, v.w * nrm);
}

// ---------------------------------------------------------------------------
// Combine: out = agg + dinv1^2*h1 (self-loop) + dinv2^2*h2 + b1 + b2, ReLU opt.
// One thread per (node, 4 feats).
// ---------------------------------------------------------------------------
__global__ __launch_bounds__(256)
void combine(const float* __restrict__ AGG,
             const float* __restrict__ H1,
             const float* __restrict__ H2,
             const float* __restrict__ dinv1,
             const float* __restrict__ dinv2,
             const float* __restrict__ b1,
             const float* __restrict__ b2,
             float* __restrict__ OUT,
             int N, int relu) {
  const int t = blockIdx.x * blockDim.x + threadIdx.x;
  if (t >= N * 32) return;
  const int v = t >> 5;
  const int q = (t & 31) * 4;
  float d1 = dinv1[v]; d1 *= d1;
  float d2 = dinv2[v]; d2 *= d2;
  const size_t off = (size_t)v * FEAT + q;
  const float4 a  = *(const float4*)(AGG + off);
  const float4 h1 = *(const float4*)(H1 + off);
  const float4 h2 = *(const float4*)(H2 + off);
  const float4 c1 = *(const float4*)(b1 + q);
  const float4 c2 = *(const float4*)(b2 + q);
  float4 o;
  o.x = a.x + d1 * h1.x + d2 * h2.x + c1.x + c2.x;
  o.y = a.y + d1 * h1.y + d2 * h2.y + c1.y + c2.y;
  o.z = a.z + d1 * h1.z + d2 * h2.z + c1.z + c2.z;
  o.w = a.w + d1 * h1.w + d2 * h2.w + c1.w + c2.w;
  if (relu) {
    o.x = fmaxf(o.x, 0.f); o.y = fmaxf(o.y, 0.f);
    o.z = fmaxf(o.z, 0.f); o.w = fmaxf(o.w, 0.f);
  }
  *(float4*)(OUT + off) = o;
}

// ---------------------------------------------------------------------------
extern "C" void kernel_launch(void* const* d_in, const int* in_sizes, int n_in,
                              void* d_out, int out_size, void* d_ws, size_t ws_size,
                              hipStream_t stream) {
  const float* x      = (const float*)d_in[0];
  const int*   ei     = (const int*)d_in[1];   // [2, E1]
  const int*   cei    = (const int*)d_in[2];   // [2, E2]
  const float* conv_w = (const float*)d_in[3]; // [D,128,128]
  const float* conv_b = (const float*)d_in[4]; // [D,128]
  const float* ctrl_w = (const float*)d_in[5];
  const float* ctrl_b = (const float*)d_in[6];

  const int N  = in_sizes[0] / FEAT;
  const int E1 = in_sizes[1] / 2;
  const int E2 = in_sizes[2] / 2;
  const int D  = in_sizes[3] / (FEAT * FEAT);

  const int* src1 = ei;        const int* dst1 = ei + E1;
  const int* src2 = cei;       const int* dst2 = cei + E2;

  const size_t NF = (size_t)N * FEAT;
  float* h1    = (float*)d_ws;
  float* h2    = h1 + NF;
  float* agg   = h2 + NF;
  float* xbuf  = agg + NF;
  float* dinv1 = xbuf + NF;
  float* dinv2 = dinv1 + N;

  const int T = 256;
  const int nodeBlocks = (N + T - 1) / T;

  // --- normalization: deg = 1 (self loop) + in-degree; dinv = rsqrt(deg) ---
  deg_init<<<nodeBlocks, T, 0, stream>>>(dinv1, dinv2, N);
  deg_count<<<(E1 + T - 1) / T, T, 0, stream>>>(dst1, dinv1, E1);
  deg_count<<<(E2 + T - 1) / T, T, 0, stream>>>(dst2, dinv2, E2);
  deg_rsqrt<<<nodeBlocks, T, 0, stream>>>(dinv1, dinv2, N);

  const int gemmBlocks = (N + 15) / 16;
  const size_t n4 = NF / 4;
  const int zeroBlocks = (int)((n4 + T - 1) / T);
  const int combBlocks = (N * 32 + T - 1) / T;

  const float* x_cur = x;
  for (int i = 0; i < D; ++i) {
    const float* W1 = conv_w + (size_t)i * FEAT * FEAT;
    const float* W2 = ctrl_w + (size_t)i * FEAT * FEAT;
    const float* B1 = conv_b + (size_t)i * FEAT;
    const float* B2 = ctrl_b + (size_t)i * FEAT;
    float* x_next = (i == D - 1) ? (float*)d_out : xbuf;

    gcn_gemm_dual_wmma<<<gemmBlocks, T, 0, stream>>>(x_cur, W1, W2, h1, h2, N);
    zero_f4<<<zeroBlocks, T, 0, stream>>>((float4*)agg, n4);
    edge_scatter<<<(E1 + 7) / 8, T, 0, stream>>>(h1, src1, dst1, dinv1, agg, E1);
    edge_scatter<<<(E2 + 7) / 8, T, 0, stream>>>(h2, src2, dst2, dinv2, agg, E2);
    combine<<<combBlocks, T, 0, stream>>>(agg, h1, h2, dinv1, dinv2, B1, B2,
                                          x_next, N, (i != D - 1) ? 1 : 0);
    x_cur = x_next;
  }
}